// BoxBlur_34505767256652
// MI455X (gfx1250) — compile-verified
//
#include <hip/hip_runtime.h>
#include <stdint.h>

typedef __attribute__((ext_vector_type(2))) float v2f;
typedef __attribute__((ext_vector_type(8))) float v8f;

#define HH 512
#define WW 512
#define NIMG 512            // B*C = 8*64
#define TILES_PER_IMG 1024  // (512/16)^2
#define PATCH 28            // 16 + 12 halo
#define PSTRIDE 28
#define WAVES_PER_WG 8

typedef __attribute__((address_space(3))) float lds_f32;

__device__ __forceinline__ int reflect_idx(int p, int n) {
    p = (p < 0) ? -p : p;             // reflect (no edge repeat); pad=6 < n-1 so one bounce suffices
    p = (p >= n) ? (2 * n - 2 - p) : p;
    return p;
}

__global__ __launch_bounds__(256) void boxblur_wmma_kernel(const float* __restrict__ in,
                                                           const float* __restrict__ kern,
                                                           float* __restrict__ out) {
    __shared__ float lds[WAVES_PER_WG][PATCH * PSTRIDE];

    const int tid  = threadIdx.x;
    const int wave = tid >> 5;
    const int lane = tid & 31;

    // one wave == one 16x16 output tile; grid covers all tiles exactly (no divergence at WMMA)
    const int tile = blockIdx.x * WAVES_PER_WG + wave;
    const int img  = tile >> 10;        // / TILES_PER_IMG
    const int rem  = tile & 1023;
    const int ty   = (rem >> 5) << 4;   // tileY * 16
    const int tx   = (rem & 31) << 4;   // tileX * 16

    const float* __restrict__ src = in + (size_t)img * (HH * WW);

    // ---- stage 1: async DMA of 28x28 halo patch directly into LDS (reflect indexing) ----
    // GLOBAL_LOAD_ASYNC_TO_LDS_B32: global -> LDS, no VGPR staging, tracked by ASYNCcnt.
    float* P = lds[wave];
    #pragma unroll 1
    for (int i = lane; i < PATCH * PATCH; i += 32) {
        const int pr = i / PATCH;
        const int pc = i - pr * PATCH;
        const int gy = reflect_idx(ty - 6 + pr, HH);
        const int gx = reflect_idx(tx - 6 + pc, WW);
        const float* gptr = src + (size_t)gy * WW + gx;
        const unsigned ldsOff = (unsigned)(uintptr_t)(lds_f32*)&P[i];  // raw LDS byte address
        asm volatile("global_load_async_to_lds_b32 %0, %1, off"
                     :
                     : "v"(ldsOff), "v"(gptr)
                     : "memory");
    }
    asm volatile("s_wait_asynccnt 0x0" ::: "memory");  // LDS writes from async path complete
    __syncthreads();

    const int   m     = lane & 15;   // row (A) / col (B,D) index within tile
    const int   h     = lane >> 4;   // lane half selects K sub-pair / row half
    const float scale = kern[0];     // uniform box kernel tap = 1/(13*13)

    // ---- stage 2: vertical 13-tap sums -> A fragments; constant banded B fragments ----
    // A (16x4 f32) layout: lanes 0-15 hold M=lane, lanes 16-31 M=lane-16;
    //   VGPR j holds K = 2*h + j within the chunk (ISA 7.12.2, 32-bit A 16x4).
    // B (4x16 f32) mirrored: lane%16 = N, VGPR j holds K-row = 2*h + j.
    v2f aF[7], bF[7];
    #pragma unroll
    for (int q = 0; q < 7; ++q) {
        #pragma unroll
        for (int j = 0; j < 2; ++j) {
            const int c = 4 * q + 2 * h + j;      // global K index in [0,28)
            float s = 0.f;
            #pragma unroll
            for (int k = 0; k < 13; ++k) s += P[(m + k) * PSTRIDE + c];
            aF[q][j] = s;                         // V[m][c] = vertical 13-tap sum
            // banded weight matrix: W[k][n] = scale iff n <= k <= n+12
            bF[q][j] = (c >= m && c <= m + 12) ? scale : 0.f;
        }
    }

    // ---- stage 3: horizontal pass as f32 WMMA, K=28 accumulated in 7 chunks of 4 ----
    v8f acc = {};
    #pragma unroll
    for (int q = 0; q < 7; ++q) {
        // 8 args: (neg_a, A, neg_b, B, c_mod, C, reuse_a, reuse_b)
        acc = __builtin_amdgcn_wmma_f32_16x16x4_f32(false, aF[q], false, bF[q],
                                                    (short)0, acc, false, false);
    }

    // ---- stage 4: store D (16x16 f32): VGPR v -> row v + 8*h, col = lane%16 ----
    float* __restrict__ dst =
        out + (size_t)img * (HH * WW) + (size_t)(ty + 8 * h) * WW + (tx + m);
    #pragma unroll
    for (int v = 0; v < 8; ++v) {
        dst[(size_t)v * WW] = acc[v];
    }
}

extern "C" void kernel_launch(void* const* d_in, const int* in_sizes, int n_in,
                              void* d_out, int out_size, void* d_ws, size_t ws_size,
                              hipStream_t stream) {
    (void)in_sizes; (void)n_in; (void)d_ws; (void)ws_size; (void)out_size;
    const float* x = (const float*)d_in[0];
    const float* k = (const float*)d_in[1];
    float*       o = (float*)d_out;

    const int total_tiles = NIMG * TILES_PER_IMG;          // 524288 wave-tiles
    dim3 grid(total_tiles / WAVES_PER_WG);                 // 65536 workgroups
    dim3 block(256);                                       // 8 waves (wave32)
    boxblur_wmma_kernel<<<grid, block, 0, stream>>>(x, k, o);
}